// ShiftedWindowMSA_41558103556671
// MI455X (gfx1250) — compile-verified
//
#include <hip/hip_runtime.h>
#include <hip/hip_bf16.h>

typedef __attribute__((ext_vector_type(2))) float v2f;
typedef __attribute__((ext_vector_type(8))) float v8f;

#define WS7    7
#define SIDE   56
#define NWIN   8
#define TOK    49
#define EMBED  96
#define HEADS  3
#define HDIM   32
#define BATCH  64
#define NROWS  (BATCH * SIDE * SIDE)   // 200704
#define QKV_N  (3 * EMBED)             // 288
#define NP     112                     // padded N per group: 2*NP % 64 == 32 (bank split)
#define QKVSZ  ((long long)BATCH * HEADS * NWIN * NWIN * TOK * HDIM)  // 19,267,584 floats

// ---- CDNA5 async global->LDS path (guarded; falls back to plain copy) ------
#if defined(__has_builtin)
#if __has_builtin(__builtin_amdgcn_global_load_async_to_lds_b128) && \
    __has_builtin(__builtin_amdgcn_s_wait_asynccnt)
#define HAVE_ASYNC_LDS 1
#endif
#endif
#ifndef HAVE_ASYNC_LDS
#define HAVE_ASYNC_LDS 0
#endif

#if HAVE_ASYNC_LDS
typedef int vsi4 __attribute__((__vector_size__(4 * sizeof(int))));
typedef __attribute__((address_space(1))) vsi4* as1_v4ptr;   // global int4*
typedef __attribute__((address_space(3))) vsi4* as3_v4ptr;   // LDS int4*
#endif

__device__ __forceinline__ void async_g2l_b128(const void* g, void* l) {
#if HAVE_ASYNC_LDS
    __builtin_amdgcn_global_load_async_to_lds_b128(
        (as1_v4ptr)(void*)g, (as3_v4ptr)l, 0, 0);
#else
    *(float4*)l = *(const float4*)g;
#endif
}

__device__ __forceinline__ void async_lds_wait() {
#if HAVE_ASYNC_LDS
    __builtin_amdgcn_s_wait_asynccnt(0);
#endif
}

__device__ __forceinline__ v8f wmma4(v2f a, v2f b, v8f c) {
    // V_WMMA_F32_16X16X4_F32: (neg_a, A, neg_b, B, c_mod, C, reuse_a, reuse_b)
    return __builtin_amdgcn_wmma_f32_16x16x4_f32(false, a, false, b, (short)0, c,
                                                 false, false);
}

// ---------------------------------------------------------------------------
// Kernel 1: QKV projection [200704x96] x [96x288] + b1, scattered into
// shifted-window layout q/k/v[(b,head,wh,ww,t,e)]. Roll(-4) folded into store.
// Block = 8 waves = 8 M-tiles x one 96-col N-group.
// w1 slice staged in LDS k-PAIRED: wsl[(p*NP + n)*2 + (k&1)], p = k/2, so each
// lane's (k,k+1) B-operand dwords are one aligned ds_load_b64 (no repacking).
// ---------------------------------------------------------------------------
__global__ __launch_bounds__(256) void qkv_gemm_scatter(
    const float* __restrict__ x, const float* __restrict__ w1,
    const float* __restrict__ b1,
    float* __restrict__ qb, float* __restrict__ kb, float* __restrict__ vb)
{
    __shared__ __align__(16) float wsl[(EMBED / 2) * NP * 2];   // 48*112*2 = 43008 B

    const int ngrp   = blockIdx.x % 3;
    const int mblock = blockIdx.x / 3;
    const int n0 = ngrp * 96;

    // stage w1[:, n0:n0+96] k-paired: thread handles (pair p, col n)
    for (int i = threadIdx.x; i < (EMBED / 2) * 96; i += 256) {
        const int p = i / 96;
        const int n = i % 96;
        float2 f2;
        f2.x = w1[(long long)(2 * p + 0) * QKV_N + n0 + n];
        f2.y = w1[(long long)(2 * p + 1) * QKV_N + n0 + n];
        *(float2*)(&wsl[(p * NP + n) * 2]) = f2;
    }
    __syncthreads();

    const int wave = threadIdx.x >> 5;
    const int lane = threadIdx.x & 31;
    const int mtile = mblock * 8 + wave;
    const int r0 = mtile * 16;

    const int mA   = lane & 15;
    const int nc   = lane & 15;
    const int kh   = (lane >> 4) * 2;
    const int half = lane >> 4;

    v8f zero = {};
    v8f c[6];
#pragma unroll
    for (int t = 0; t < 6; ++t) c[t] = zero;

    const float* arow = x + (long long)(r0 + mA) * EMBED;
#pragma unroll 4
    for (int k0 = 0; k0 < EMBED; k0 += 4) {
        v2f a = *(const v2f*)(arow + k0 + kh);
        const float* bp = &wsl[((k0 / 2 + half) * NP + nc) * 2];
#pragma unroll
        for (int t = 0; t < 6; ++t)
            c[t] = wmma4(a, *(const v2f*)(bp + t * 32), c[t]);
    }

#pragma unroll
    for (int r = 0; r < 8; ++r) {
        const int gr  = r0 + r + 8 * half;
        const int b_  = gr / (SIDE * SIDE);
        const int rem = gr % (SIDE * SIDE);
        const int s1 = rem / SIDE, s2 = rem % SIDE;
        const int i1 = (s1 + SIDE - 4) % SIDE;    // roll(-4): dest index
        const int i2 = (s2 + SIDE - 4) % SIDE;
        const int wh = i1 / WS7, m1 = i1 % WS7;
        const int ww = i2 / WS7, m2 = i2 % WS7;
        const int tok = m1 * WS7 + m2;
        const long long rowbase =
            ((((long long)(b_ * HEADS) * NWIN + wh) * NWIN + ww) * TOK + tok) * HDIM;
#pragma unroll
        for (int t = 0; t < 6; ++t) {
            const int gc = n0 + t * 16 + nc;      // 0..287
            const int ch = gc / 3;                // channel (K fastest in reshape)
            const int which = gc % 3;             // 0=q 1=k 2=v
            const int head = ch / HDIM;
            const int e = ch % HDIM;
            const float val = c[t][r] + b1[gc];
            const long long idx =
                rowbase + (long long)head * (NWIN * NWIN * TOK * HDIM) + e;
            float* dst = (which == 0) ? qb : ((which == 1) ? kb : vb);
            dst[idx] = val;
        }
    }
}

// ---------------------------------------------------------------------------
// Kernel 2: per-window attention. Block = 4 waves = one (b,head,wh,ww).
// S = Q K^T / sqrt(32) + bias (+ shift masks), softmax, O = P V.
// Padded 49 -> 64; wave w owns score rows [16w,16w+16).
// Q/K/V staged via GLOBAL_LOAD_ASYNC_TO_LDS_B128 when available.
// ---------------------------------------------------------------------------
__global__ __launch_bounds__(128) void window_attention(
    const float* __restrict__ qb, const float* __restrict__ kb,
    const float* __restrict__ vb, const float* __restrict__ rel_bias,
    float* __restrict__ att_out)
{
    const int inst = blockIdx.x;                  // (b*3+head)*64 + wh*8+ww
    const int winf = inst & 63;
    const int bh   = inst >> 6;
    const int wh = winf >> 3, ww = winf & 7;
    const int head = bh % HEADS, b_ = bh / HEADS;

    __shared__ __align__(16) float qs[TOK * HDIM];
    __shared__ __align__(16) float ks[TOK * HDIM];
    __shared__ __align__(16) float vs[TOK * HDIM];
    __shared__ __align__(16) float bias_s[TOK * TOK];
    __shared__ __align__(16) float ps[4][16 * 64];   // per-wave P stripe

    const long long base = (long long)inst * (TOK * HDIM);
    {
        const float4* gq = (const float4*)(qb + base);
        const float4* gk = (const float4*)(kb + base);
        const float4* gv = (const float4*)(vb + base);
        for (int i = threadIdx.x; i < (TOK * HDIM) / 4; i += 128) {
            async_g2l_b128(gq + i, (float4*)qs + i);
            async_g2l_b128(gk + i, (float4*)ks + i);
            async_g2l_b128(gv + i, (float4*)vs + i);
        }
    }
    for (int i = threadIdx.x; i < TOK * TOK; i += 128) bias_s[i] = rel_bias[i];
    async_lds_wait();
    __syncthreads();

    const int wave = threadIdx.x >> 5;
    const int lane = threadIdx.x & 31;
    const int m0   = wave * 16;
    const int half = lane >> 4;
    const int nc   = lane & 15;
    const int kh   = half * 2;
    const int mA   = m0 + (lane & 15);

    v2f zero2 = {};
    v8f zero = {};
    v8f sacc[4];
#pragma unroll
    for (int t = 0; t < 4; ++t) sacc[t] = zero;

    // S = Q * K^T   (K-dim = 32); (k,k+1) contiguous in qs/ks rows -> b64 loads
#pragma unroll
    for (int k0 = 0; k0 < HDIM; k0 += 4) {
        v2f a = (mA < TOK) ? *(const v2f*)(qs + mA * HDIM + k0 + kh) : zero2;
#pragma unroll
        for (int t = 0; t < 4; ++t) {
            const int nB = t * 16 + nc;
            v2f b = (nB < TOK) ? *(const v2f*)(ks + nB * HDIM + k0 + kh) : zero2;
            sacc[t] = wmma4(a, b, sacc[t]);
        }
    }

    const float scale = 0.17677669529663687f;     // 1/sqrt(32)
    const bool lastrow = (wh == NWIN - 1);
    const bool lastcol = (ww == NWIN - 1);

    // softmax per score row (rows striped: vgpr r holds row m0+r+8*half)
#pragma unroll
    for (int r = 0; r < 8; ++r) {
        const int m = m0 + r + 8 * half;
        float sv[4];
        float rowmax = -1e30f;
#pragma unroll
        for (int t = 0; t < 4; ++t) {
            const int n = t * 16 + nc;
            float s = sacc[t][r] * scale;
            if (m < TOK && n < TOK) {
                s += bias_s[m * TOK + n];
                if (lastrow && ((m >= 28) != (n >= 28)))           s = -1e30f;
                if (lastcol && (((m % 7) >= 4) != ((n % 7) >= 4))) s = -1e30f;
            } else {
                s = -1e30f;                        // padded column
            }
            sv[t] = s;
            rowmax = fmaxf(rowmax, s);
        }
        rowmax = fmaxf(rowmax, __shfl_xor(rowmax, 1, 32));
        rowmax = fmaxf(rowmax, __shfl_xor(rowmax, 2, 32));
        rowmax = fmaxf(rowmax, __shfl_xor(rowmax, 4, 32));
        rowmax = fmaxf(rowmax, __shfl_xor(rowmax, 8, 32));
        float rsum = 0.0f;
#pragma unroll
        for (int t = 0; t < 4; ++t) {
            const float e = __expf(sv[t] - rowmax);
            sv[t] = e;
            rsum += e;
        }
        rsum += __shfl_xor(rsum, 1, 32);
        rsum += __shfl_xor(rsum, 2, 32);
        rsum += __shfl_xor(rsum, 4, 32);
        rsum += __shfl_xor(rsum, 8, 32);
        const float inv = 1.0f / rsum;
#pragma unroll
        for (int t = 0; t < 4; ++t)
            ps[wave][(r + 8 * half) * 64 + t * 16 + nc] = sv[t] * inv;
    }
    // wave reads only its own ps stripe -> no block barrier needed

    // O = P * V   (K-dim = 64 padded, padded P cols are ~0 and V guarded)
    v8f oacc[2];
    oacc[0] = zero; oacc[1] = zero;
#pragma unroll
    for (int k0 = 0; k0 < 64; k0 += 4) {
        v2f a = *(const v2f*)(&ps[wave][(lane & 15) * 64 + k0 + kh]);
        const int kr = k0 + kh;
#pragma unroll
        for (int t = 0; t < 2; ++t) {
            const int e = t * 16 + nc;
            v2f b;
            b.x = (kr     < TOK) ? vs[(kr    ) * HDIM + e] : 0.0f;
            b.y = (kr + 1 < TOK) ? vs[(kr + 1) * HDIM + e] : 0.0f;
            oacc[t] = wmma4(a, b, oacc[t]);
        }
    }

    // scatter O with window-merge + roll(+3) folded into indexing
#pragma unroll
    for (int r = 0; r < 8; ++r) {
        const int m = m0 + r + 8 * half;
        if (m >= TOK) continue;
        const int m1 = m / WS7, m2 = m % WS7;
        const int s1 = (wh * WS7 + m1 + 3) % SIDE;
        const int s2 = (ww * WS7 + m2 + 3) % SIDE;
        const long long row = ((long long)b_ * SIDE + s1) * SIDE + s2;
#pragma unroll
        for (int t = 0; t < 2; ++t) {
            const int e = t * 16 + nc;
            att_out[row * EMBED + head * HDIM + e] = oacc[t][r];
        }
    }
}

// ---------------------------------------------------------------------------
// Kernel 3: output projection [200704x96] x [96x96] + b2 -> d_out.
// Block = 8 waves = 8 M-tiles, all six N-tiles; w2 staged k-paired in LDS.
// ---------------------------------------------------------------------------
__global__ __launch_bounds__(256) void proj_gemm(
    const float* __restrict__ a_in, const float* __restrict__ w2,
    const float* __restrict__ b2, float* __restrict__ out)
{
    __shared__ __align__(16) float wsl[(EMBED / 2) * NP * 2];   // 43008 B

    for (int i = threadIdx.x; i < (EMBED / 2) * 96; i += 256) {
        const int p = i / 96;
        const int n = i % 96;
        float2 f2;
        f2.x = w2[(long long)(2 * p + 0) * EMBED + n];
        f2.y = w2[(long long)(2 * p + 1) * EMBED + n];
        *(float2*)(&wsl[(p * NP + n) * 2]) = f2;
    }
    __syncthreads();

    const int wave = threadIdx.x >> 5;
    const int lane = threadIdx.x & 31;
    const int mtile = blockIdx.x * 8 + wave;      // 12544 total, exact
    const int r0 = mtile * 16;

    const int mA   = lane & 15;
    const int nc   = lane & 15;
    const int kh   = (lane >> 4) * 2;
    const int half = lane >> 4;

    v8f zero = {};
    v8f c[6];
#pragma unroll
    for (int t = 0; t < 6; ++t) c[t] = zero;

    const float* arow = a_in + (long long)(r0 + mA) * EMBED;
#pragma unroll 4
    for (int k0 = 0; k0 < EMBED; k0 += 4) {
        v2f a = *(const v2f*)(arow + k0 + kh);
        const float* bp = &wsl[((k0 / 2 + half) * NP + nc) * 2];
#pragma unroll
        for (int t = 0; t < 6; ++t)
            c[t] = wmma4(a, *(const v2f*)(bp + t * 32), c[t]);
    }

#pragma unroll
    for (int r = 0; r < 8; ++r) {
        const long long gr = r0 + r + 8 * half;
#pragma unroll
        for (int t = 0; t < 6; ++t) {
            const int gc = t * 16 + nc;
            out[gr * EMBED + gc] = c[t][r] + b2[gc];
        }
    }
}

extern "C" void kernel_launch(void* const* d_in, const int* in_sizes, int n_in,
                              void* d_out, int out_size, void* d_ws, size_t ws_size,
                              hipStream_t stream) {
    const float* x        = (const float*)d_in[0];
    const float* w1       = (const float*)d_in[1];
    const float* b1       = (const float*)d_in[2];
    const float* w2       = (const float*)d_in[3];
    const float* b2       = (const float*)d_in[4];
    const float* rel_bias = (const float*)d_in[5];
    float* out = (float*)d_out;

    float* qb  = (float*)d_ws;
    float* kb  = qb + QKVSZ;
    float* vb  = kb + QKVSZ;
    float* att = vb + QKVSZ;

    // K1: 1568 m-blocks * 3 n-groups
    qkv_gemm_scatter<<<1568 * 3, 256, 0, stream>>>(x, w1, b1, qb, kb, vb);
    // K2: 64 batch * 3 heads * 64 windows = 12288 blocks
    window_attention<<<12288, 128, 0, stream>>>(qb, kb, vb, rel_bias, att);
    // K3: 12544 mtiles / 8 waves = 1568 blocks
    proj_gemm<<<1568, 256, 0, stream>>>(att, w2, b2, out);
}